// MultiHeadAttentionRel_74646531604662
// MI455X (gfx1250) — compile-verified
//
#include <hip/hip_runtime.h>
#include <hip/hip_bf16.h>
#include <math.h>

// ---------------------------------------------------------------------------
// Transformer-XL relative attention, B=1 S=512 H=768 NH=12 D=64 (all fp32).
//
// Refactor: BD = einsum('bhqd,bqkhd->bhqk', q+v, pos@Wr^T) is re-associated as
//   W2[h,q,e] = sum_d (q+v)[h,q,d] * Wr[h*64+d, e]          (0.6 GFLOP)
//   BD[h,q,k] = sum_e W2[h,q,e] * pos[q,k,e] + (q+v)·br     (4.8 GFLOP)
// so the 309-GFLOP pos-projection disappears and pos (805 MB) is streamed
// exactly once -> HBM-bound at ~35us @ 23.3 TB/s. Needed f32 throughput while
// streaming (~140 TFLOP/s) fits under the V_WMMA_F32_16X16X4_F32 ceiling, so
// the whole pipeline stays in f32 WMMA (reference precision, no downcast).
// ---------------------------------------------------------------------------

typedef __attribute__((ext_vector_type(2))) float v2f;
typedef __attribute__((ext_vector_type(8))) float v8f;

// D = A(16x4) * B(4x16) + C ; f32, wave32
// A layout: lane L holds M = L&15; K = vgpr + 2*(L>=16)
// B layout: lane L holds N = L&15; K = vgpr + 2*(L>=16)
// C layout: lane L holds N = L&15; M = vgpr + 8*(L>=16)
__device__ __forceinline__ v8f wmma_f32(v2f a, v2f b, v8f c) {
  return __builtin_amdgcn_wmma_f32_16x16x4_f32(false, a, false, b,
                                               (short)0, c, false, false);
}

#define Sdim 512
#define Hdim 768
#define NHd  12
#define Dd   64

// -------- C[M,N] = A[M,K] @ W[N,K]^T + bias[N]  (q/k/v projections) --------
__global__ void gemm_bias_t_kernel(const float* __restrict__ A,
                                   const float* __restrict__ W,
                                   const float* __restrict__ bias,
                                   float* __restrict__ C,
                                   int M, int N, int K) {
  const int lane = threadIdx.x;
  const int m0   = (blockIdx.y * blockDim.y + threadIdx.y) * 16;
  const int n0   = blockIdx.x * 16;
  const int mrow = m0 + (lane & 15);
  const int nrow = n0 + (lane & 15);
  const int koff = (lane >> 4) * 2;

  v8f acc = {};
  for (int k = 0; k < K; k += 4) {
    v2f a, b;
    a.x = A[(size_t)mrow * K + k + koff];
    a.y = A[(size_t)mrow * K + k + koff + 1];
    b.x = W[(size_t)nrow * K + k + koff];      // B[e,n] = W[n,e]
    b.y = W[(size_t)nrow * K + k + koff + 1];
    acc = wmma_f32(a, b, acc);
  }
  const int cm = m0 + 8 * (lane >> 4);
  const int cn = n0 + (lane & 15);
  const float bv = bias[cn];
#pragma unroll
  for (int r = 0; r < 8; ++r)
    C[(size_t)(cm + r) * N + cn] = acc[r] + bv;
}

// -------- W2[h,q,e] = sum_d (qp[q,h*64+d] + vbias[h,d]) * Wr[h*64+d, e] ----
__global__ void w2_kernel(const float* __restrict__ qp,
                          const float* __restrict__ Wr,
                          const float* __restrict__ vbias,
                          float* __restrict__ W2) {
  const int lane = threadIdx.x;
  const int h    = blockIdx.z;
  const int q0   = (blockIdx.y * blockDim.y + threadIdx.y) * 16;
  const int e0   = blockIdx.x * 16;
  const int mrow = q0 + (lane & 15);
  const int ecol = e0 + (lane & 15);
  const int koff = (lane >> 4) * 2;

  v8f acc = {};
  for (int d = 0; d < Dd; d += 4) {
    const int dd = d + koff;
    v2f a, b;
    a.x = qp[(size_t)mrow * Hdim + h * Dd + dd]     + vbias[h * Dd + dd];
    a.y = qp[(size_t)mrow * Hdim + h * Dd + dd + 1] + vbias[h * Dd + dd + 1];
    b.x = Wr[(size_t)(h * Dd + dd)     * Hdim + ecol];
    b.y = Wr[(size_t)(h * Dd + dd + 1) * Hdim + ecol];
    acc = wmma_f32(a, b, acc);
  }
  const int cm = q0 + 8 * (lane >> 4);
#pragma unroll
  for (int r = 0; r < 8; ++r)
    W2[((size_t)h * Sdim + cm + r) * Hdim + ecol] = acc[r];
}

// -------- cq[h,q] = sum_d (qp[q,h*64+d] + vbias[h,d]) * br[h*64+d] ---------
__global__ void cq_kernel(const float* __restrict__ qp,
                          const float* __restrict__ vbias,
                          const float* __restrict__ br,
                          float* __restrict__ cq) {
  const int idx = blockIdx.x * blockDim.x + threadIdx.x;
  if (idx >= NHd * Sdim) return;
  const int h = idx / Sdim, q = idx % Sdim;
  float s = 0.f;
#pragma unroll 8
  for (int d = 0; d < Dd; ++d)
    s += (qp[(size_t)q * Hdim + h * Dd + d] + vbias[h * Dd + d]) * br[h * Dd + d];
  cq[idx] = s;
}

// -------- score[h,q,k] = sum_d (qp+u)[q,h,d] * kp[k,h,d] (content term) ----
__global__ void ac_kernel(const float* __restrict__ qp,
                          const float* __restrict__ kp,
                          const float* __restrict__ ubias,
                          float* __restrict__ score) {
  const int lane = threadIdx.x;
  const int h    = blockIdx.z;
  const int q0   = (blockIdx.y * blockDim.y + threadIdx.y) * 16;
  const int k0   = blockIdx.x * 16;
  const int mrow = q0 + (lane & 15);
  const int kcol = k0 + (lane & 15);
  const int koff = (lane >> 4) * 2;

  v8f acc = {};
  for (int d = 0; d < Dd; d += 4) {
    const int dd = d + koff;
    v2f a, b;
    a.x = qp[(size_t)mrow * Hdim + h * Dd + dd]     + ubias[h * Dd + dd];
    a.y = qp[(size_t)mrow * Hdim + h * Dd + dd + 1] + ubias[h * Dd + dd + 1];
    b.x = kp[(size_t)kcol * Hdim + h * Dd + dd];
    b.y = kp[(size_t)kcol * Hdim + h * Dd + dd + 1];
    acc = wmma_f32(a, b, acc);
  }
  const int cm = q0 + 8 * (lane >> 4);
#pragma unroll
  for (int r = 0; r < 8; ++r)
    score[((size_t)h * Sdim + cm + r) * Sdim + kcol] = acc[r];
}

// -------- BD stream: score = (score + W2[:,q,:]@pos[q]^T + cq)/8, mask -----
// One block per (ktile-group, q). pos is streamed exactly once from HBM.
// W2[:,q,:] (12x768 = 36 KB) staged in LDS, shared by 8 waves (8 k-tiles).
__global__ void __launch_bounds__(256)
bd_kernel(const float* __restrict__ W2, const float* __restrict__ pos,
          const float* __restrict__ cq, const int* __restrict__ key_mask,
          float* __restrict__ score) {
  __shared__ float sW2[NHd * Hdim];
  const int q   = blockIdx.y;
  const int tid = threadIdx.y * 32 + threadIdx.x;
  for (int i = tid; i < NHd * Hdim; i += 256) {
    const int h = i / Hdim, e = i % Hdim;
    sW2[i] = W2[((size_t)h * Sdim + q) * Hdim + e];
  }
  __syncthreads();

  const int lane = threadIdx.x;
  const int k0   = (blockIdx.x * 8 + threadIdx.y) * 16;
  const int kcol = k0 + (lane & 15);
  const int koff = (lane >> 4) * 2;
  const int m    = lane & 15;                       // A-row = head (pad to 16)
  const float* pcol = pos + ((size_t)q * Sdim + kcol) * Hdim;

  v8f acc = {};
  for (int e = 0; e < Hdim; e += 4) {
    const int ee = e + koff;
    if ((e & 63) == 0)
      __builtin_prefetch(pcol + e + 512, 0, 1);     // stream-ahead, 2 KB
    v2f a, b;
    a.x = (m < NHd) ? sW2[m * Hdim + ee]     : 0.f;
    a.y = (m < NHd) ? sW2[m * Hdim + ee + 1] : 0.f;
    b.x = pcol[ee];                                 // B[e,k] = pos[q,k,e]
    b.y = pcol[ee + 1];
    acc = wmma_f32(a, b, acc);
  }

  const int hbase = 8 * (lane >> 4);
  const int mk    = key_mask[kcol];
#pragma unroll
  for (int r = 0; r < 8; ++r) {
    const int h = hbase + r;
    if (h < NHd) {
      const size_t si = ((size_t)h * Sdim + q) * Sdim + kcol;
      const float s = (score[si] + acc[r] + cq[h * Sdim + q]) * 0.125f;
      score[si] = (mk == 0) ? -1e15f : s;
    }
  }
}

// -------- softmax over k (row length 512), one block per (h,q) row ---------
__global__ void softmax_kernel(float* __restrict__ score) {
  __shared__ float redm[8], reds[8];
  float* s = score + (size_t)blockIdx.x * Sdim;
  const int tid = threadIdx.x;                      // 256 threads, 2 elem each
  float v0 = s[tid], v1 = s[tid + 256];

  float m = fmaxf(v0, v1);
  for (int off = 16; off > 0; off >>= 1) m = fmaxf(m, __shfl_xor(m, off, 32));
  if ((tid & 31) == 0) redm[tid >> 5] = m;
  __syncthreads();
  float rmax = redm[0];
#pragma unroll
  for (int i = 1; i < 8; ++i) rmax = fmaxf(rmax, redm[i]);

  const float e0 = __expf(v0 - rmax), e1 = __expf(v1 - rmax);
  float sum = e0 + e1;
  for (int off = 16; off > 0; off >>= 1) sum += __shfl_xor(sum, off, 32);
  if ((tid & 31) == 0) reds[tid >> 5] = sum;
  __syncthreads();
  float rsum = 0.f;
#pragma unroll
  for (int i = 0; i < 8; ++i) rsum += reds[i];

  const float inv = 1.0f / rsum;
  s[tid] = e0 * inv;
  s[tid + 256] = e1 * inv;
}

// -------- out[q, h*64+d] = sum_k attn[h,q,k] * vp[k, h*64+d] ---------------
__global__ void out_kernel(const float* __restrict__ attn,
                           const float* __restrict__ vp,
                           float* __restrict__ out) {
  const int lane = threadIdx.x;
  const int h    = blockIdx.z;
  const int q0   = (blockIdx.y * blockDim.y + threadIdx.y) * 16;
  const int d0   = blockIdx.x * 16;
  const int mrow = q0 + (lane & 15);
  const int dcol = d0 + (lane & 15);
  const int koff = (lane >> 4) * 2;
  const float* arow = attn + ((size_t)h * Sdim + mrow) * Sdim;

  v8f acc = {};
  for (int k = 0; k < Sdim; k += 4) {
    const int kk = k + koff;
    v2f a, b;
    a.x = arow[kk];
    a.y = arow[kk + 1];
    b.x = vp[(size_t)kk       * Hdim + h * Dd + dcol];
    b.y = vp[(size_t)(kk + 1) * Hdim + h * Dd + dcol];
    acc = wmma_f32(a, b, acc);
  }
  const int cm = q0 + 8 * (lane >> 4);
#pragma unroll
  for (int r = 0; r < 8; ++r)
    out[(size_t)(cm + r) * Hdim + h * Dd + dcol] = acc[r];
}

// ---------------------------------------------------------------------------
extern "C" void kernel_launch(void* const* d_in, const int* in_sizes, int n_in,
                              void* d_out, int out_size, void* d_ws, size_t ws_size,
                              hipStream_t stream) {
  const float* key    = (const float*)d_in[0];
  const float* query  = (const float*)d_in[1];
  const float* value  = (const float*)d_in[2];
  const float* pos    = (const float*)d_in[3];
  const int*   kmask  = (const int*)  d_in[4];
  const float* Wk     = (const float*)d_in[5];
  const float* bk     = (const float*)d_in[6];
  const float* Wq     = (const float*)d_in[7];
  const float* bq     = (const float*)d_in[8];
  const float* Wv     = (const float*)d_in[9];
  const float* bv     = (const float*)d_in[10];
  const float* Wr     = (const float*)d_in[11];
  const float* br     = (const float*)d_in[12];
  const float* ubias  = (const float*)d_in[13];
  const float* vbias  = (const float*)d_in[14];
  float* out = (float*)d_out;

  // workspace carve-up (floats): qp|kp|vp|W2|cq|score  ~ 34.6 MB total
  float* ws    = (float*)d_ws;
  float* qp    = ws;                               // 512*768
  float* kp    = qp + Sdim * Hdim;
  float* vp    = kp + Sdim * Hdim;
  float* W2    = vp + Sdim * Hdim;                 // 12*512*768
  float* cq    = W2 + (size_t)NHd * Sdim * Hdim;   // 12*512
  float* score = cq + NHd * Sdim;                  // 12*512*512

  const dim3 blk(32, 4);
  // q/k/v projections: [512,768] @ [768,768]^T + bias
  gemm_bias_t_kernel<<<dim3(Hdim / 16, Sdim / 64), blk, 0, stream>>>(query, Wq, bq, qp, Sdim, Hdim, Hdim);
  gemm_bias_t_kernel<<<dim3(Hdim / 16, Sdim / 64), blk, 0, stream>>>(key,   Wk, bk, kp, Sdim, Hdim, Hdim);
  gemm_bias_t_kernel<<<dim3(Hdim / 16, Sdim / 64), blk, 0, stream>>>(value, Wv, bv, vp, Sdim, Hdim, Hdim);

  // W2 = (q+v_bias) @ Wr (per-head), and its br dot term
  w2_kernel<<<dim3(Hdim / 16, Sdim / 64, NHd), blk, 0, stream>>>(qp, Wr, vbias, W2);
  cq_kernel<<<dim3((NHd * Sdim + 255) / 256), dim3(256), 0, stream>>>(qp, vbias, br, cq);

  // content term AC -> score
  ac_kernel<<<dim3(Sdim / 16, Sdim / 64, NHd), blk, 0, stream>>>(qp, kp, ubias, score);

  // position term: stream pos once; score = (AC+BD+cq)/8, masked
  bd_kernel<<<dim3(Sdim / 128, Sdim), dim3(32, 8), 0, stream>>>(W2, pos, cq, kmask, score);

  // softmax over k
  softmax_kernel<<<dim3(NHd * Sdim), dim3(256), 0, stream>>>(score);

  // out = attn @ v
  out_kernel<<<dim3(Dd / 16, Sdim / 64, NHd), blk, 0, stream>>>(score, vp, out);
}